// yolov8_GradCamLoss_49400713839278
// MI455X (gfx1250) — compile-verified
//
#include <hip/hip_runtime.h>

// Sum-reduction of three fp32 tensors (YOLOv8 head): ~225.8 MB streamed once.
// Bandwidth-bound: floor ~9.7us at 23.3 TB/s. Strategy:
//   - grid-stride B128 non-temporal loads (read-once; don't thrash 192MB L2)
//   - accumulate via V_WMMA_F32_16X16X4_F32 with B = ones (rowsum trick);
//     summation is permutation-invariant so any lane->matrix-slot mapping works
//   - accumulator holds 16x the sum (rowsum replicated over 16 cols) -> *1/16
//   - NO bounds check in the hot loop: full iterations are unchecked (block
//     counts chosen so stride divides n4 exactly for the real shapes); one
//     guarded tail iteration for generality. EXEC is all-ones at every WMMA.
//   - manual 4x unroll, 2 independent accumulators: 4 B128 loads in flight,
//     two independent WMMA dependency chains.
//   - deterministic: per-block partials to d_ws, fixed-order single-block finish

typedef __attribute__((ext_vector_type(2))) float v2f;
typedef __attribute__((ext_vector_type(4))) float v4f;
typedef __attribute__((ext_vector_type(8))) float v8f;

__device__ __forceinline__ v8f wmma_rowsum2(v8f c, v4f v, v2f ones) {
    // Fold one float4 (two 16x4 A-tile slots) into the accumulator.
    v2f a0 = {v.x, v.y};
    v2f a1 = {v.z, v.w};
    c = __builtin_amdgcn_wmma_f32_16x16x4_f32(false, a0, false, ones, (short)0, c, false, false);
    c = __builtin_amdgcn_wmma_f32_16x16x4_f32(false, a1, false, ones, (short)0, c, false, false);
    return c;
}

__device__ __forceinline__ float wave_reduce(float s) {
    // wave32 butterfly reduction
    #pragma unroll
    for (int off = 16; off > 0; off >>= 1)
        s += __shfl_xor(s, off, 32);
    return s;
}

__global__ __launch_bounds__(256) void sum_partial_wmma(
    const v4f* __restrict__ in, int n4, float* __restrict__ partials)
{
    const int tid    = blockIdx.x * blockDim.x + threadIdx.x;
    const int stride = gridDim.x * blockDim.x;
    const int iters  = n4 / stride;        // full, unchecked iterations
    const int iters4 = iters & ~3;

    v8f acc0 = {};                         // two independent 16x16 f32 accs
    v8f acc1 = {};
    const v2f ones = {1.0f, 1.0f};         // B matrix = all ones (4x16)

    int base = tid;

    // Main loop: 4 independent NT B128 loads in flight, 2 WMMA chains.
    for (int i = 0; i < iters4; i += 4) {
        v4f v0 = __builtin_nontemporal_load(in + base);
        v4f v1 = __builtin_nontemporal_load(in + base + stride);
        v4f v2 = __builtin_nontemporal_load(in + base + 2 * stride);
        v4f v3 = __builtin_nontemporal_load(in + base + 3 * stride);
        base += 4 * stride;
        acc0 = wmma_rowsum2(acc0, v0, ones);
        acc1 = wmma_rowsum2(acc1, v1, ones);
        acc0 = wmma_rowsum2(acc0, v2, ones);
        acc1 = wmma_rowsum2(acc1, v3, ones);
    }

    // Leftover full iterations (0..3).
    for (int i = iters4; i < iters; ++i) {
        v4f v = __builtin_nontemporal_load(in + base);
        base += stride;
        acc0 = wmma_rowsum2(acc0, v, ones);
    }

    // Ragged tail (n4 % stride): zero for the chosen launch configs, kept for
    // generality. Divergent load re-converges before the WMMA issues.
    if (iters * stride < n4) {
        const int idx = tid + iters * stride;
        v4f v = {0.0f, 0.0f, 0.0f, 0.0f};
        if (idx < n4) v = __builtin_nontemporal_load(in + idx);
        acc1 = wmma_rowsum2(acc1, v, ones);
    }

    // Sum this lane's 16 accumulator slots, then reduce across the wave.
    float s = (acc0[0] + acc1[0]) + (acc0[1] + acc1[1]) +
              (acc0[2] + acc1[2]) + (acc0[3] + acc1[3]) +
              (acc0[4] + acc1[4]) + (acc0[5] + acc1[5]) +
              (acc0[6] + acc1[6]) + (acc0[7] + acc1[7]);
    s = wave_reduce(s);

    __shared__ float wsum[8];
    const int wave = threadIdx.x >> 5;
    if ((threadIdx.x & 31) == 0) wsum[wave] = s;
    __syncthreads();

    if (threadIdx.x == 0) {
        float t = 0.0f;
        #pragma unroll
        for (int w = 0; w < 8; ++w) t += wsum[w];
        // accumulator counted each element 16x (rowsum broadcast over 16 cols)
        partials[blockIdx.x] = t * 0.0625f;
    }
}

__global__ __launch_bounds__(256) void sum_final(
    const float* __restrict__ partials, int nPartials,
    const float* __restrict__ x0, int n0,
    const float* __restrict__ x1, int n1,
    const float* __restrict__ x2, int n2,
    float* __restrict__ out)
{
    float s = 0.0f;
    for (int i = threadIdx.x; i < nPartials; i += 256)
        s += partials[i];

    // Ragged %4 tails (zero for these shapes, handled for generality).
    if (threadIdx.x == 0) {
        for (int i = (n0 & ~3); i < n0; ++i) s += x0[i];
        for (int i = (n1 & ~3); i < n1; ++i) s += x1[i];
        for (int i = (n2 & ~3); i < n2; ++i) s += x2[i];
    }

    s = wave_reduce(s);

    __shared__ float wsum[8];
    const int wave = threadIdx.x >> 5;
    if ((threadIdx.x & 31) == 0) wsum[wave] = s;
    __syncthreads();

    if (threadIdx.x == 0) {
        float t = 0.0f;
        #pragma unroll
        for (int w = 0; w < 8; ++w) t += wsum[w];
        out[0] = t;
    }
}

extern "C" void kernel_launch(void* const* d_in, const int* in_sizes, int n_in,
                              void* d_out, int out_size, void* d_ws, size_t ws_size,
                              hipStream_t stream) {
    const float* x0 = (const float*)d_in[0];
    const float* x1 = (const float*)d_in[1];
    const float* x2 = (const float*)d_in[2];
    const int n0 = in_sizes[0];   // 43,008,000
    const int n1 = in_sizes[1];   // 10,752,000
    const int n2 = in_sizes[2];   //  2,688,000

    float* partials = (float*)d_ws;

    // Block counts chosen so stride = blocks*256 divides n/4 exactly:
    //   n0/4 = 10,752,000 = 840*256*50  -> 50 full iterations, no tail
    //   n1/4 =  2,688,000 = 210*256*50  -> 50 full iterations, no tail
    //   n2/4 =    672,000 = 105*256*25  -> 25 full iterations, no tail
    // 1155 blocks x 8 wave32s ~ 9.2K waves: ample to saturate 23.3 TB/s.
    const int B0 = 840, B1 = 210, B2 = 105;

    sum_partial_wmma<<<B0, 256, 0, stream>>>((const v4f*)x0, n0 / 4, partials);
    sum_partial_wmma<<<B1, 256, 0, stream>>>((const v4f*)x1, n1 / 4, partials + B0);
    sum_partial_wmma<<<B2, 256, 0, stream>>>((const v4f*)x2, n2 / 4, partials + B0 + B1);

    sum_final<<<1, 256, 0, stream>>>(partials, B0 + B1 + B2,
                                     x0, n0, x1, n1, x2, n2,
                                     (float*)d_out);
}